// MHSAOneBlock_3289944949305
// MI455X (gfx1250) — compile-verified
//
#include <hip/hip_runtime.h>
#include <hip/hip_bf16.h>
#include <math.h>

typedef __attribute__((ext_vector_type(2))) float v2f;
typedef __attribute__((ext_vector_type(8))) float v8f;

#define D_MODEL 1024
#define INNER   1024
#define NHEAD   16
#define HDIM    64
#define TLEN    2048
#define BATCH   2
#define MROWS   (BATCH * TLEN)   // 4096
#define FFDIM   4096
#define NVOCAB  32000
#define KBLK    64

__device__ __forceinline__ v8f wmma4(v2f a, v2f b, v8f c) {
  // V_WMMA_F32_16X16X4_F32 : full fp32 matrix op
  return __builtin_amdgcn_wmma_f32_16x16x4_f32(false, a, false, b, (short)0, c,
                                               false, false);
}

__device__ __forceinline__ float gelu_exact(float x) {
  return 0.5f * x * (1.0f + erff(x * 0.70710678118654752f));
}

// ---------------------------------------------------------------- embed + pos
__global__ void embed_kernel(const int* __restrict__ x,
                             const float* __restrict__ emb,
                             const float* __restrict__ pos,
                             float* __restrict__ h) {
  int row = blockIdx.x;                 // b*T + t
  int t = row & (TLEN - 1);
  int tok = x[row];
  const float* e = emb + (size_t)tok * D_MODEL;
  const float* p = pos + (size_t)t * D_MODEL;
  float* o = h + (size_t)row * D_MODEL;
  for (int i = threadIdx.x; i < D_MODEL; i += blockDim.x) o[i] = e[i] + p[i];
}

// ---------------------------------------------------------------- layernorm
__global__ __launch_bounds__(256) void ln_kernel(const float* __restrict__ in,
                                                 const float* __restrict__ g,
                                                 const float* __restrict__ b,
                                                 float* __restrict__ out) {
  int row = blockIdx.x;
  const float* xr = in + (size_t)row * D_MODEL;
  float* orow = out + (size_t)row * D_MODEL;
  float s = 0.f, s2 = 0.f;
  for (int i = threadIdx.x; i < D_MODEL; i += blockDim.x) {
    float v = xr[i];
    s += v; s2 += v * v;
  }
  __shared__ float sh0[256];
  __shared__ float sh1[256];
  sh0[threadIdx.x] = s; sh1[threadIdx.x] = s2;
  __syncthreads();
  for (int o = 128; o > 0; o >>= 1) {
    if ((int)threadIdx.x < o) {
      sh0[threadIdx.x] += sh0[threadIdx.x + o];
      sh1[threadIdx.x] += sh1[threadIdx.x + o];
    }
    __syncthreads();
  }
  float mu = sh0[0] * (1.0f / D_MODEL);
  float var = sh1[0] * (1.0f / D_MODEL) - mu * mu;
  float rstd = rsqrtf(var + 1e-5f);
  for (int i = threadIdx.x; i < D_MODEL; i += blockDim.x)
    orow[i] = (xr[i] - mu) * rstd * g[i] + b[i];
}

// ---------------------------------------------------------------- WMMA GEMM
// C[M,N] = A[M,K] @ W[K,N] (+bias) (+gelu) (+res)
// Block = 256 threads = 8 waves. Block computes a 32-row M panel; the A panel
// is staged through LDS per 64-wide K block (coalesced float4 loads, shared by
// all 8 waves). Each wave computes 2 M-tiles x 4 N-tiles = 8 WMMAs per K-step,
// reusing each B fragment twice. W panel prefetched one K-block ahead.
template <bool BIAS, bool GELU, bool RES>
__global__ __launch_bounds__(256) void gemm_kernel(
    const float* __restrict__ A, const float* __restrict__ W,
    const float* __restrict__ bias, const float* __restrict__ res,
    float* __restrict__ C, int M, int N, int K) {
  __shared__ float As[32 * KBLK];       // 8 KB

  int tid = threadIdx.x;
  int lane = tid & 31;
  int wave = tid >> 5;
  int nl = lane & 15;
  int khalf = lane >> 4;

  int mt = blockIdx.x;                  // 32-row panel index
  int nTiles = N >> 4;
  int ntBase = (blockIdx.y * 8 + wave) * 4;
  bool active = ntBase < nTiles;        // wave-uniform
  int ncol = active ? ntBase * 16 + nl : nl;   // clamp inactive to valid addr

  const float* apanel = A + (size_t)(mt * 32) * K;

  // cooperative A staging indices: thread t -> row t/8, cols (t%8)*8 .. +7
  int smrow = tid >> 3;
  int skcol = (tid & 7) * 8;

  v8f acc[2][4] = {};
  for (int k0 = 0; k0 < K; k0 += KBLK) {
    {
      const float* s = apanel + (size_t)smrow * K + k0 + skcol;
      float4 v0 = *(const float4*)(s);
      float4 v1 = *(const float4*)(s + 4);
      *(float4*)(&As[smrow * KBLK + skcol]) = v0;
      *(float4*)(&As[smrow * KBLK + skcol + 4]) = v1;
    }
    __syncthreads();
    if (k0 + KBLK < K) {  // pull next W panel toward the caches
      __builtin_prefetch(W + (size_t)(k0 + KBLK + 2 * khalf) * N + ncol, 0, 1);
      __builtin_prefetch(W + (size_t)(k0 + KBLK + 32 + 2 * khalf) * N + ncol, 0, 1);
    }
#pragma unroll 4
    for (int k = 0; k < KBLK; k += 4) {
      v2f a0, a1;
      a0.x = As[nl * KBLK + k + 2 * khalf];
      a0.y = As[nl * KBLK + k + 2 * khalf + 1];
      a1.x = As[(16 + nl) * KBLK + k + 2 * khalf];
      a1.y = As[(16 + nl) * KBLK + k + 2 * khalf + 1];
      const float* wk = W + (size_t)(k0 + k + 2 * khalf) * N + ncol;
#pragma unroll
      for (int j = 0; j < 4; ++j) {
        v2f b; b.x = wk[j * 16]; b.y = wk[(size_t)N + j * 16];
        acc[0][j] = wmma4(a0, b, acc[0][j]);
        acc[1][j] = wmma4(a1, b, acc[1][j]);
      }
    }
    __syncthreads();
  }

  if (!active) return;
  // epilogue: C layout — VGPR r -> row base + r + 8*khalf, col = ncol + j*16
#pragma unroll
  for (int mi = 0; mi < 2; ++mi) {
#pragma unroll
    for (int r = 0; r < 8; ++r) {
      int grow = mt * 32 + mi * 16 + r + 8 * khalf;
#pragma unroll
      for (int j = 0; j < 4; ++j) {
        int col = ncol + j * 16;
        float v = acc[mi][j][r];
        if (BIAS) v += bias[col];
        if (GELU) v = gelu_exact(v);
        size_t idx = (size_t)grow * N + col;
        if (RES) v += res[idx];
        C[idx] = v;
      }
    }
  }
}

// ---------------------------------------------------------------- attention
// One wave per (b, h, q-tile of 16 rows). Flash-attention online softmax.
__global__ __launch_bounds__(32) void attn_kernel(const float* __restrict__ qkv,
                                                  float* __restrict__ aout) {
  int qt = blockIdx.x;     // 0..T/16-1
  int hh = blockIdx.y;     // head
  int bb = blockIdx.z;     // batch
  int lane = threadIdx.x;
  int nl = lane & 15;
  int khalf = lane >> 4;
  const int LDQ = 3 * INNER;   // 3072

  const float* base = qkv + (size_t)bb * TLEN * LDQ;
  const float* qrow = base + (size_t)(qt * 16 + nl) * LDQ + hh * HDIM;
  const float* kbase = base + INNER + hh * HDIM;
  const float* vbase = base + 2 * INNER + hh * HDIM;

  // Q fragment in registers: A-operand layout for 16 K-steps of 4
  v2f qa[16];
#pragma unroll
  for (int s = 0; s < 16; ++s) {
    qa[s].x = qrow[4 * s + 2 * khalf];
    qa[s].y = qrow[4 * s + 2 * khalf + 1];
  }

  v8f o[4] = {};
  float m[8], l[8];
#pragma unroll
  for (int r = 0; r < 8; ++r) { m[r] = -__builtin_inff(); l[r] = 0.f; }

  __shared__ float pbuf[16 * 16];

  for (int kt = 0; kt <= qt; ++kt) {
    // S = Q @ K^T  (B operand: N = key index, element = Key[key, k])
    v8f s = {};
    const float* krow = kbase + (size_t)(kt * 16 + nl) * LDQ + 2 * khalf;
#pragma unroll
    for (int st = 0; st < 16; ++st) {
      v2f bf; bf.x = krow[4 * st]; bf.y = krow[4 * st + 1];
      s = wmma4(qa[st], bf, s);
    }
    // scale + causal mask + online softmax
    float p[8];
#pragma unroll
    for (int r = 0; r < 8; ++r) {
      float v = s[r] * 0.125f;   // 1/sqrt(64)
      if (kt == qt && nl > r + 8 * khalf) v = -__builtin_inff();
      float rowmax = v;
      for (int off = 8; off > 0; off >>= 1)
        rowmax = fmaxf(rowmax, __shfl_xor(rowmax, off, 16));
      float mnew = fmaxf(m[r], rowmax);
      float pv = __expf(v - mnew);
      float rowsum = pv;
      for (int off = 8; off > 0; off >>= 1)
        rowsum += __shfl_xor(rowsum, off, 16);
      float f = __expf(m[r] - mnew);
      l[r] = l[r] * f + rowsum;
      m[r] = mnew;
#pragma unroll
      for (int j = 0; j < 4; ++j) o[j][r] *= f;
      p[r] = pv;
    }
    // stage P through LDS to reshape C-layout -> A-operand layout
#pragma unroll
    for (int r = 0; r < 8; ++r) pbuf[(r + 8 * khalf) * 16 + nl] = p[r];
    __syncthreads();
    // O += P @ V
#pragma unroll
    for (int kk = 0; kk < 4; ++kk) {
      v2f pa;
      pa.x = pbuf[nl * 16 + 4 * kk + 2 * khalf];
      pa.y = pbuf[nl * 16 + 4 * kk + 2 * khalf + 1];
      const float* vrow = vbase + (size_t)(kt * 16 + 4 * kk + 2 * khalf) * LDQ + nl;
#pragma unroll
      for (int j = 0; j < 4; ++j) {
        v2f bf; bf.x = vrow[j * 16]; bf.y = vrow[LDQ + j * 16];
        o[j] = wmma4(pa, bf, o[j]);
      }
    }
    __syncthreads();
  }

  // normalize and write (B,T,INNER) at head offset
  float* orow = aout + (size_t)bb * TLEN * INNER + hh * HDIM;
#pragma unroll
  for (int r = 0; r < 8; ++r) {
    int gr = qt * 16 + r + 8 * khalf;
    float inv = 1.0f / l[r];
#pragma unroll
    for (int j = 0; j < 4; ++j)
      orow[(size_t)gr * INNER + j * 16 + nl] = o[j][r] * inv;
  }
}

// ---------------------------------------------------------------- launch
extern "C" void kernel_launch(void* const* d_in, const int* in_sizes, int n_in,
                              void* d_out, int out_size, void* d_ws,
                              size_t ws_size, hipStream_t stream) {
  const int*   x      = (const int*)  d_in[0];
  const float* emb    = (const float*)d_in[1];
  const float* pos    = (const float*)d_in[2];
  const float* w_qkv  = (const float*)d_in[3];
  const float* w_out  = (const float*)d_in[4];
  const float* ln1_g  = (const float*)d_in[5];
  const float* ln1_b  = (const float*)d_in[6];
  const float* ln2_g  = (const float*)d_in[7];
  const float* ln2_b  = (const float*)d_in[8];
  const float* w_ffn1 = (const float*)d_in[9];
  const float* b_ffn1 = (const float*)d_in[10];
  const float* w_ffn2 = (const float*)d_in[11];
  const float* b_ffn2 = (const float*)d_in[12];
  const float* w_head = (const float*)d_in[13];
  const float* b_head = (const float*)d_in[14];
  float* logits = (float*)d_out;

  float* h   = (float*)d_ws;                       // 4096 x 1024
  float* h1  = h   + (size_t)MROWS * D_MODEL;      // 4096 x 1024 (ln1 / ln2)
  float* qkv = h1  + (size_t)MROWS * D_MODEL;      // 4096 x 3072
  float* att = qkv + (size_t)MROWS * 3 * INNER;    // 4096 x 1024
  float* ff  = att + (size_t)MROWS * INNER;        // 4096 x 4096

  dim3 blk(256);
  // N columns covered per block = 8 waves * 4 tiles * 16 = 512
  auto gy = [](int n) { return (n / 16 + 31) / 32; };

  // h = emb[x] + pos
  embed_kernel<<<MROWS, blk, 0, stream>>>(x, emb, pos, h);
  // h1 = LN1(h)
  ln_kernel<<<MROWS, blk, 0, stream>>>(h, ln1_g, ln1_b, h1);
  // qkv = h1 @ w_qkv
  gemm_kernel<false, false, false><<<dim3(MROWS / 32, gy(3 * INNER)), blk, 0,
      stream>>>(h1, w_qkv, nullptr, nullptr, qkv, MROWS, 3 * INNER, D_MODEL);
  // attention
  attn_kernel<<<dim3(TLEN / 16, NHEAD, BATCH), dim3(32), 0, stream>>>(qkv, att);
  // h = h + att @ w_out
  gemm_kernel<false, false, true><<<dim3(MROWS / 32, gy(D_MODEL)), blk, 0,
      stream>>>(att, w_out, nullptr, h, h, MROWS, D_MODEL, INNER);
  // h1 = LN2(h)
  ln_kernel<<<MROWS, blk, 0, stream>>>(h, ln2_g, ln2_b, h1);
  // ff = gelu(h1 @ w_ffn1 + b_ffn1)
  gemm_kernel<true, true, false><<<dim3(MROWS / 32, gy(FFDIM)), blk, 0,
      stream>>>(h1, w_ffn1, b_ffn1, nullptr, ff, MROWS, FFDIM, D_MODEL);
  // h = h + ff @ w_ffn2 + b_ffn2
  gemm_kernel<true, false, true><<<dim3(MROWS / 32, gy(D_MODEL)), blk, 0,
      stream>>>(ff, w_ffn2, b_ffn2, h, h, MROWS, D_MODEL, FFDIM);
  // logits = h @ w_head + b_head (2000 N-tiles -> grid.y=63, uniform guard)
  gemm_kernel<true, false, false><<<dim3(MROWS / 32, gy(NVOCAB)), blk, 0,
      stream>>>(h, w_head, b_head, nullptr, logits, MROWS, NVOCAB, D_MODEL);
}